// LogTGN_78941498901088
// MI455X (gfx1250) — compile-verified
//
#include <hip/hip_runtime.h>

// ---- problem dims ----
#define N_NODES 500000
#define N_SUB   50000
#define NE      500000
#define NB      5000
#define DD      100     // memory/embedding dim
#define TDIM    100     // time dim
#define MSGD    100
#define KEDGE   200     // TD + MSG
#define NT_T    7       // N tiles of 16 covering 100 (pad to 112)

// ---- WMMA f32 16x16x4 ----
typedef __attribute__((ext_vector_type(2))) float v2f;
typedef __attribute__((ext_vector_type(8))) float v8f;

__device__ __forceinline__ v8f wmma4(v2f a, v2f b, v8f c) {
    // D = A(16x4) x B(4x16) + C, fp32, wave32
    return __builtin_amdgcn_wmma_f32_16x16x4_f32(
        false, a, false, b, (short)0, c, false, false);
}

// ordered-uint key for float atomic max
__device__ __forceinline__ unsigned fkey(float f) {
    unsigned b = __float_as_uint(f);
    return (b & 0x80000000u) ? ~b : (b | 0x80000000u);
}
__device__ __forceinline__ float funkey(unsigned k) {
    unsigned b = (k & 0x80000000u) ? (k & 0x7FFFFFFFu) : ~k;
    return __uint_as_float(b);
}

// ---------------- K1: gather memory rows, last_update, zero attn state ----
__global__ void k_gather(const float* __restrict__ mem,
                         const int* __restrict__ lastup,
                         const int* __restrict__ nid,
                         float* __restrict__ z,
                         int* __restrict__ lu_sub,
                         unsigned* __restrict__ mmax,
                         float* __restrict__ denom) {
    int i = blockIdx.x * blockDim.x + threadIdx.x;
    if (i < N_SUB * DD) {
        int row = i / DD, col = i % DD;
        z[i] = mem[(long)nid[row] * DD + col];
    }
    if (i < N_SUB) lu_sub[i] = lastup[nid[i]];
    if (i < N_SUB * 2) { mmax[i] = 0u; denom[i] = 0.f; }
}

// ---------------- K2: rel_t per edge -------------------------------------
__global__ void k_relt(const int* __restrict__ lu_sub,
                       const int* __restrict__ src,
                       const int* __restrict__ t,
                       float* __restrict__ rel_t) {
    int e = blockIdx.x * blockDim.x + threadIdx.x;
    if (e < NE) rel_t[e] = (float)(lu_sub[src[e]] - t[e]);
}

// ---------------- K3: q/k/v/skip GEMMs via WMMA f32 ----------------------
// one wave per (matrix, M-tile); 7 N-tile accumulators; K = 100 (25 steps)
__global__ __launch_bounds__(256) void k_node_gemm(
    const float* __restrict__ z,
    const float* __restrict__ Wq, const float* __restrict__ bq,
    const float* __restrict__ Wk, const float* __restrict__ bk,
    const float* __restrict__ Wv, const float* __restrict__ bv,
    const float* __restrict__ Ws, const float* __restrict__ bs,
    float* __restrict__ q, float* __restrict__ k,
    float* __restrict__ v, float* __restrict__ outb) {
    const int MT = N_SUB / 16;                       // 3125
    int wave = blockIdx.x * (blockDim.x >> 5) + (threadIdx.x >> 5);
    if (wave >= 4 * MT) return;                      // wave-uniform
    int mat = wave / MT, mt = wave % MT;
    const float *W, *bias; float* dst;
    if      (mat == 0) { W = Wq; bias = bq; dst = q;    }
    else if (mat == 1) { W = Wk; bias = bk; dst = k;    }
    else if (mat == 2) { W = Wv; bias = bv; dst = v;    }
    else               { W = Ws; bias = bs; dst = outb; }

    int lane = threadIdx.x & 31, half = lane >> 4, lr = lane & 15;
    int m0 = mt * 16;
    const float* arow = z + (long)(m0 + lr) * DD;

    v8f acc[NT_T] = {};
    for (int kk = 0; kk < 25; ++kk) {
        int kb = kk * 4 + 2 * half;
        v2f a; a.x = arow[kb]; a.y = arow[kb + 1];
        #pragma unroll
        for (int nt = 0; nt < NT_T; ++nt) {
            int n = nt * 16 + lr;
            v2f bb;
            bb.x = (n < DD) ? W[(kb)     * DD + n] : 0.f;
            bb.y = (n < DD) ? W[(kb + 1) * DD + n] : 0.f;
            acc[nt] = wmma4(a, bb, acc[nt]);
        }
    }
    #pragma unroll
    for (int nt = 0; nt < NT_T; ++nt) {
        int n = nt * 16 + lr;
        if (n < DD) {
            float bn = bias[n];
            #pragma unroll
            for (int r = 0; r < 8; ++r) {
                int m = m0 + r + 8 * half;
                dst[(long)m * DD + n] = acc[nt][r] + bn;
            }
        }
    }
}

// ---------------- K4: edge-feature GEMM, edge_attr built on the fly ------
// e[E,100] = concat(cos(rel_t*w_time+b_time), msg) @ We[200,100]
__global__ __launch_bounds__(256) void k_edge_gemm(
    const float* __restrict__ rel_t, const float* __restrict__ msg,
    const float* __restrict__ w_time, const float* __restrict__ b_time,
    const float* __restrict__ We, float* __restrict__ ef) {
    const int MT = NE / 16;                          // 31250
    int wave = blockIdx.x * (blockDim.x >> 5) + (threadIdx.x >> 5);
    if (wave >= MT) return;
    int lane = threadIdx.x & 31, half = lane >> 4, lr = lane & 15;
    int m0 = wave * 16;
    int mrow = m0 + lr;
    float rt = rel_t[mrow];
    const float* mrow_msg = msg + (long)mrow * MSGD;

    v8f acc[NT_T] = {};
    for (int kk = 0; kk < 50; ++kk) {                // K = 200
        int kb = kk * 4 + 2 * half;
        v2f a;
        #pragma unroll
        for (int r = 0; r < 2; ++r) {
            int kidx = kb + r;
            float av = (kidx < TDIM)
                ? __cosf(0.f) * 0.f + cosf(rt * w_time[kidx] + b_time[kidx])
                : mrow_msg[kidx - TDIM];
            if (r == 0) a.x = av; else a.y = av;
        }
        #pragma unroll
        for (int nt = 0; nt < NT_T; ++nt) {
            int n = nt * 16 + lr;
            v2f bb;
            bb.x = (n < DD) ? We[(kb)     * DD + n] : 0.f;
            bb.y = (n < DD) ? We[(kb + 1) * DD + n] : 0.f;
            acc[nt] = wmma4(a, bb, acc[nt]);
        }
    }
    #pragma unroll
    for (int nt = 0; nt < NT_T; ++nt) {
        int n = nt * 16 + lr;
        if (n < DD) {
            #pragma unroll
            for (int r = 0; r < 8; ++r) {
                int m = m0 + r + 8 * half;
                ef[(long)m * DD + n] = acc[nt][r];
            }
        }
    }
}

// ---------------- K5: alpha + segment max (ordered-uint atomic) ----------
__global__ void k_alpha(const float* __restrict__ q, const float* __restrict__ k,
                        const float* __restrict__ ef,
                        const int* __restrict__ src, const int* __restrict__ dst,
                        float* __restrict__ alpha, unsigned* __restrict__ mmax) {
    int tid = blockIdx.x * blockDim.x + threadIdx.x;
    if (tid >= NE * 2) return;
    int e = tid >> 1, h = tid & 1;
    int s = src[e], d = dst[e];
    const float* qp = q  + (long)d * DD + h * 50;
    const float* kp = k  + (long)s * DD + h * 50;
    const float* ep = ef + (long)e * DD + h * 50;
    float sum = 0.f;
    #pragma unroll 5
    for (int c = 0; c < 50; ++c) sum += qp[c] * (kp[c] + ep[c]);
    float al = sum * 0.14142135623730950488f;        // 1/sqrt(50)
    alpha[tid] = al;
    atomicMax(&mmax[d * 2 + h], fkey(al));
}

// ---------------- K6: exp(alpha - max), segment sum ----------------------
__global__ void k_norm(const int* __restrict__ dst,
                       const unsigned* __restrict__ mmax,
                       float* __restrict__ alpha, float* __restrict__ denom) {
    int tid = blockIdx.x * blockDim.x + threadIdx.x;
    if (tid >= NE * 2) return;
    int e = tid >> 1, h = tid & 1;
    int d = dst[e];
    float m = funkey(mmax[d * 2 + h]);
    float a = expf(alpha[tid] - m);
    alpha[tid] = a;
    atomicAdd(&denom[d * 2 + h], a);
}

// ---------------- K7: weighted aggregation into skip-initialized out -----
__global__ void k_agg(const int* __restrict__ src, const int* __restrict__ dst,
                      const float* __restrict__ v, const float* __restrict__ ef,
                      const float* __restrict__ alpha, const float* __restrict__ denom,
                      float* __restrict__ outb) {
    int tid = blockIdx.x * blockDim.x + threadIdx.x;
    if (tid >= NE * DD) return;
    int e = tid / DD, j = tid % DD, h = j / 50;
    int d = dst[e];
    float a = alpha[e * 2 + h] / (denom[d * 2 + h] + 1e-16f);
    float val = v[(long)src[e] * DD + j] + ef[(long)e * DD + j];
    atomicAdd(&outb[(long)d * DD + j], a * val);
}

// ---------------- K8: h = relu(out[si]@Wps + out[di]@Wpd + biases) -------
__global__ __launch_bounds__(256) void k_pred_h(
    const float* __restrict__ outb,
    const int* __restrict__ si, const int* __restrict__ di,
    const float* __restrict__ Wps, const float* __restrict__ bps,
    const float* __restrict__ Wpd, const float* __restrict__ bpd,
    float* __restrict__ hbuf) {
    const int MT = (NB + 15) / 16;                   // 313
    int wave = blockIdx.x * (blockDim.x >> 5) + (threadIdx.x >> 5);
    if (wave >= MT) return;
    int lane = threadIdx.x & 31, half = lane >> 4, lr = lane & 15;
    int m0 = wave * 16;
    int m = m0 + lr;
    bool mv = (m < NB);
    const float* as = outb + (long)(mv ? si[m] : 0) * DD;
    const float* ad = outb + (long)(mv ? di[m] : 0) * DD;

    v8f acc[NT_T] = {};
    for (int kk = 0; kk < 25; ++kk) {                // src-half
        int kb = kk * 4 + 2 * half;
        v2f a;
        a.x = mv ? as[kb]     : 0.f;
        a.y = mv ? as[kb + 1] : 0.f;
        #pragma unroll
        for (int nt = 0; nt < NT_T; ++nt) {
            int n = nt * 16 + lr;
            v2f bb;
            bb.x = (n < DD) ? Wps[(kb)     * DD + n] : 0.f;
            bb.y = (n < DD) ? Wps[(kb + 1) * DD + n] : 0.f;
            acc[nt] = wmma4(a, bb, acc[nt]);
        }
    }
    for (int kk = 0; kk < 25; ++kk) {                // dst-half, chained C
        int kb = kk * 4 + 2 * half;
        v2f a;
        a.x = mv ? ad[kb]     : 0.f;
        a.y = mv ? ad[kb + 1] : 0.f;
        #pragma unroll
        for (int nt = 0; nt < NT_T; ++nt) {
            int n = nt * 16 + lr;
            v2f bb;
            bb.x = (n < DD) ? Wpd[(kb)     * DD + n] : 0.f;
            bb.y = (n < DD) ? Wpd[(kb + 1) * DD + n] : 0.f;
            acc[nt] = wmma4(a, bb, acc[nt]);
        }
    }
    #pragma unroll
    for (int nt = 0; nt < NT_T; ++nt) {
        int n = nt * 16 + lr;
        if (n < DD) {
            float bn = bps[n] + bpd[n];
            #pragma unroll
            for (int r = 0; r < 8; ++r) {
                int mm = m0 + r + 8 * half;
                if (mm < NB) {
                    float x = acc[nt][r] + bn;
                    hbuf[(long)mm * DD + n] = x > 0.f ? x : 0.f;
                }
            }
        }
    }
}

// ---------------- K9: pred = h @ Wpf + bpf -------------------------------
__global__ __launch_bounds__(256) void k_pred_out(
    const float* __restrict__ hbuf,
    const float* __restrict__ Wpf, const float* __restrict__ bpf,
    float* __restrict__ pred) {
    const int MT = (NB + 15) / 16;
    int wave = blockIdx.x * (blockDim.x >> 5) + (threadIdx.x >> 5);
    if (wave >= MT) return;
    int lane = threadIdx.x & 31, half = lane >> 4, lr = lane & 15;
    int m0 = wave * 16;
    int m = m0 + lr;
    bool mv = (m < NB);
    const float* arow = hbuf + (long)(mv ? m : 0) * DD;

    v8f acc[NT_T] = {};
    for (int kk = 0; kk < 25; ++kk) {
        int kb = kk * 4 + 2 * half;
        v2f a;
        a.x = mv ? arow[kb]     : 0.f;
        a.y = mv ? arow[kb + 1] : 0.f;
        #pragma unroll
        for (int nt = 0; nt < NT_T; ++nt) {
            int n = nt * 16 + lr;
            v2f bb;
            bb.x = (n < DD) ? Wpf[(kb)     * DD + n] : 0.f;
            bb.y = (n < DD) ? Wpf[(kb + 1) * DD + n] : 0.f;
            acc[nt] = wmma4(a, bb, acc[nt]);
        }
    }
    #pragma unroll
    for (int nt = 0; nt < NT_T; ++nt) {
        int n = nt * 16 + lr;
        if (n < DD) {
            float bn = bpf[n];
            #pragma unroll
            for (int r = 0; r < 8; ++r) {
                int mm = m0 + r + 8 * half;
                if (mm < NB) pred[(long)mm * DD + n] = acc[nt][r] + bn;
            }
        }
    }
}

// ---------------- launch ---------------------------------------------------
extern "C" void kernel_launch(void* const* d_in, const int* in_sizes, int n_in,
                              void* d_out, int out_size, void* d_ws, size_t ws_size,
                              hipStream_t stream) {
    const float* memory      = (const float*)d_in[0];
    const int*   last_update = (const int*)  d_in[1];
    const int*   n_id        = (const int*)  d_in[2];
    const int*   edge_index  = (const int*)  d_in[3];
    const int*   src         = edge_index;
    const int*   dstE        = edge_index + NE;
    const int*   t           = (const int*)  d_in[4];
    const float* msg         = (const float*)d_in[5];
    const int*   src_idx     = (const int*)  d_in[6];
    const int*   dst_idx     = (const int*)  d_in[7];
    const float* w_time      = (const float*)d_in[8];
    const float* b_time      = (const float*)d_in[9];
    const float* Wq = (const float*)d_in[10]; const float* bq = (const float*)d_in[11];
    const float* Wk = (const float*)d_in[12]; const float* bk = (const float*)d_in[13];
    const float* Wv = (const float*)d_in[14]; const float* bv = (const float*)d_in[15];
    const float* We = (const float*)d_in[16];
    const float* Ws = (const float*)d_in[17]; const float* bs = (const float*)d_in[18];
    const float* Wps = (const float*)d_in[19]; const float* bps = (const float*)d_in[20];
    const float* Wpd = (const float*)d_in[21]; const float* bpd = (const float*)d_in[22];
    const float* Wpf = (const float*)d_in[23]; const float* bpf = (const float*)d_in[24];
    float* pred = (float*)d_out;

    // workspace layout (float offsets)
    float*    ws     = (float*)d_ws;
    float*    z      = ws;                          //  5,000,000
    int*      lu_sub = (int*)(ws + 5000000);        //     50,000
    float*    rel_t  = ws + 5050000;                //    500,000
    float*    q      = ws + 5550000;                //  5,000,000
    float*    k      = ws + 10550000;               //  5,000,000
    float*    v      = ws + 15550000;               //  5,000,000
    float*    outb   = ws + 20550000;               //  5,000,000
    float*    ef     = ws + 25550000;               // 50,000,000
    float*    alpha  = ws + 75550000;               //  1,000,000
    unsigned* mmax   = (unsigned*)(ws + 76550000);  //    100,000
    float*    denom  = ws + 76650000;               //    100,000
    float*    hbuf   = ws + 76750000;               //    500,000

    const int B = 256;
    // 1) gather + zero attention state
    k_gather<<<(N_SUB * DD + B - 1) / B, B, 0, stream>>>(
        memory, last_update, n_id, z, lu_sub, mmax, denom);
    // 2) rel_t
    k_relt<<<(NE + B - 1) / B, B, 0, stream>>>(lu_sub, src, t, rel_t);
    // 3) node GEMMs (q,k,v,skip->outb): 4*3125 waves
    k_node_gemm<<<(4 * (N_SUB / 16) + 7) / 8, B, 0, stream>>>(
        z, Wq, bq, Wk, bk, Wv, bv, Ws, bs, q, k, v, outb);
    // 4) edge GEMM: 31250 waves
    k_edge_gemm<<<((NE / 16) + 7) / 8, B, 0, stream>>>(
        rel_t, msg, w_time, b_time, We, ef);
    // 5) attention score + segment max
    k_alpha<<<(NE * 2 + B - 1) / B, B, 0, stream>>>(q, k, ef, src, dstE, alpha, mmax);
    // 6) exp + segment sum
    k_norm<<<(NE * 2 + B - 1) / B, B, 0, stream>>>(dstE, mmax, alpha, denom);
    // 7) weighted aggregation into outb (skip already there)
    k_agg<<<(NE * DD + B - 1) / B, B, 0, stream>>>(src, dstE, v, ef, alpha, denom, outb);
    // 8) predictor hidden
    k_pred_h<<<(((NB + 15) / 16) + 7) / 8, B, 0, stream>>>(
        outb, src_idx, dst_idx, Wps, bps, Wpd, bpd, hbuf);
    // 9) predictor output
    k_pred_out<<<(((NB + 15) / 16) + 7) / 8, B, 0, stream>>>(hbuf, Wpf, bpf, pred);
}